// GridEncoder_54778012893616
// MI455X (gfx1250) — compile-verified
//
#include <hip/hip_runtime.h>
#include <stdint.h>

// ---------------------------------------------------------------------------
// Instant-NGP style multiresolution hash-grid encoder for MI455X (gfx1250).
// D=3, L=16 levels, C=2 features, H=16 base res, per-level scale 2.0.
// Levels 0..2 are dense-grid indexed (no modulo needed: side^3 <= table size),
// levels 3..15 are hashed with table size 2^19 (modulo == mask).
// Level-0 table (38.4 KB) is staged into LDS via the Tensor Data Mover.
// ---------------------------------------------------------------------------

#define NLEVELS 16
#define TAB0    4920        // level-0 entries (17^3 rounded up to 8)
#define HASH_MASK 0x7FFFFu  // 2^19 - 1

typedef __attribute__((ext_vector_type(4))) unsigned int u32x4;
typedef __attribute__((ext_vector_type(8))) int          i32x8;
typedef __attribute__((ext_vector_type(4))) int          i32x4;
typedef __attribute__((ext_vector_type(4))) float        f32x4;

// Per-level entry offsets into the embedding table (in float2 entries).
__device__ constexpr uint32_t kOff[NLEVELS] = {
    0u,       4920u,    40864u,   315496u,
    839784u,  1364072u, 1888360u, 2412648u,
    2936936u, 3461224u, 3985512u, 4509800u,
    5034088u, 5558376u, 6082664u, 6606952u
};

__global__ __launch_bounds__(256)
void grid_encoder_kernel(const float* __restrict__ xin,
                         const float* __restrict__ emb,
                         float* __restrict__ out,
                         int B)
{
    __shared__ float s_tab[TAB0 * 2];   // 39360 bytes

    // ---- Stage level-0 table into LDS with the Tensor Data Mover ---------
    // One wave per block issues the DMA; TENSORcnt tracks completion.
    if (threadIdx.x < 32u) {
        uint64_t ga = (uint64_t)(uintptr_t)emb;           // global tile start
        uint32_t la = (uint32_t)(uintptr_t)(&s_tab[0]);   // LDS byte offset

        // D# group 0: count=1 (valid, load), lds_addr, 57-bit global addr, type=2
        u32x4 g0;
        g0.x = 1u;
        g0.y = la;
        g0.z = (uint32_t)(ga & 0xFFFFFFFFu);
        g0.w = (uint32_t)(ga >> 32) | (2u << 30);

        // D# group 1: data_size=3 (8 bytes/elem), 1-D tile of TAB0 elements.
        //  [17:16] data_size | [79:48] tensor_dim0 | [111:80] tensor_dim1
        //  [127:112] tile_dim0 | [143:128] tile_dim1 | [207:160] dim0 stride
        i32x8 g1;
        g1[0] = (int)(3u << 16);                 // data_size = 8B
        g1[1] = (int)((uint32_t)TAB0 << 16);     // tensor_dim0[15:0]
        g1[2] = (int)(1u << 16);                 // tensor_dim0 hi=0, tensor_dim1=1
        g1[3] = (int)((uint32_t)TAB0 << 16);     // tile_dim0 = TAB0
        g1[4] = 1;                               // tile_dim1 = 1, tile_dim2 = 0
        g1[5] = TAB0;                            // tensor_dim0_stride low
        g1[6] = 0;
        g1[7] = 0;

        i32x4 z4 = {0, 0, 0, 0};                 // groups 2/3 unused (<=2D)
        i32x8 z8 = {0, 0, 0, 0, 0, 0, 0, 0};
        __builtin_amdgcn_tensor_load_to_lds(g0, g1, z4, z4, z8, 0);
        __builtin_amdgcn_s_wait_tensorcnt(0);
    }
    __syncthreads();

    const int stride_t = (int)(gridDim.x * blockDim.x);

    for (int b = (int)(blockIdx.x * blockDim.x + threadIdx.x); b < B; b += stride_t) {
        // Inputs are read once: nontemporal so they don't evict the table in L2.
        const float px = __builtin_nontemporal_load(xin + 3 * (size_t)b + 0);
        const float py = __builtin_nontemporal_load(xin + 3 * (size_t)b + 1);
        const float pz = __builtin_nontemporal_load(xin + 3 * (size_t)b + 2);

        float acc[2 * NLEVELS];

        #pragma unroll
        for (int l = 0; l < NLEVELS; ++l) {
            const float scale = (float)((16u << l) - 1u);

            const float fx = px * scale + 0.5f;
            const float fy = py * scale + 0.5f;
            const float fz = pz * scale + 0.5f;
            const float gx = floorf(fx), gy = floorf(fy), gz = floorf(fz);
            const float wx1 = fx - gx, wy1 = fy - gy, wz1 = fz - gz;
            const float wx0 = 1.0f - wx1, wy0 = 1.0f - wy1, wz0 = 1.0f - wz1;
            const uint32_t ux = (uint32_t)gx;
            const uint32_t uy = (uint32_t)gy;
            const uint32_t uz = (uint32_t)gz;

            // yz weight products (shared across the x pairs)
            const float w00 = wy0 * wz0, w10 = wy1 * wz0;
            const float w01 = wy0 * wz1, w11 = wy1 * wz1;

            float a0 = 0.0f, a1 = 0.0f;

            auto gmem = [&a0, &a1](const float2* t, uint32_t idx, float w) {
                float2 e = t[idx];                       // global_load_b64 gather
                a0 = fmaf(w, e.x, a0);
                a1 = fmaf(w, e.y, a1);
            };

            if (l == 0) {
                // Dense 17^3 grid, served from LDS (ds_load gathers).
                // s_tab has static storage duration: no capture needed.
                const uint32_t base = ux + uy * 17u + uz * 289u;
                auto gl = [&a0, &a1](uint32_t idx, float w) {
                    const float e0 = s_tab[2u * idx];
                    const float e1 = s_tab[2u * idx + 1u];
                    a0 = fmaf(w, e0, a0);
                    a1 = fmaf(w, e1, a1);
                };
                gl(base,        wx0 * w00);  gl(base + 1u,   wx1 * w00);
                gl(base + 17u,  wx0 * w10);  gl(base + 18u,  wx1 * w10);
                gl(base + 289u, wx0 * w01);  gl(base + 290u, wx1 * w01);
                gl(base + 306u, wx0 * w11);  gl(base + 307u, wx1 * w11);
            } else if (l < 3) {
                // Dense grid levels: linear index, provably < table size.
                const uint32_t side = (l == 1) ? 33u : 65u;
                const uint32_t s2   = side * side;
                const float2* tab   = (const float2*)emb + kOff[l];
                const uint32_t base = ux + uy * side + uz * s2;
                gmem(tab, base,             wx0 * w00);
                gmem(tab, base + 1u,        wx1 * w00);
                gmem(tab, base + side,      wx0 * w10);
                gmem(tab, base + side + 1u, wx1 * w10);
                gmem(tab, base + s2,             wx0 * w01);
                gmem(tab, base + s2 + 1u,        wx1 * w01);
                gmem(tab, base + s2 + side,      wx0 * w11);
                gmem(tab, base + s2 + side + 1u, wx1 * w11);
            } else {
                // Hashed levels: 2^19 entries -> modulo is a mask.
                const float2* tab = (const float2*)emb + kOff[l];
                const uint32_t hy0 = uy * 2654435761u, hy1 = hy0 + 2654435761u;
                const uint32_t hz0 = uz * 805459861u,  hz1 = hz0 + 805459861u;
                const uint32_t x0 = ux, x1 = ux + 1u;
                gmem(tab, (x0 ^ hy0 ^ hz0) & HASH_MASK, wx0 * w00);
                gmem(tab, (x1 ^ hy0 ^ hz0) & HASH_MASK, wx1 * w00);
                gmem(tab, (x0 ^ hy1 ^ hz0) & HASH_MASK, wx0 * w10);
                gmem(tab, (x1 ^ hy1 ^ hz0) & HASH_MASK, wx1 * w10);
                gmem(tab, (x0 ^ hy0 ^ hz1) & HASH_MASK, wx0 * w01);
                gmem(tab, (x1 ^ hy0 ^ hz1) & HASH_MASK, wx1 * w01);
                gmem(tab, (x0 ^ hy1 ^ hz1) & HASH_MASK, wx0 * w11);
                gmem(tab, (x1 ^ hy1 ^ hz1) & HASH_MASK, wx1 * w11);
            }

            acc[2 * l]     = a0;
            acc[2 * l + 1] = a1;
        }

        // Each thread owns a contiguous 128B output row: 8 NT b128 stores,
        // every lane fully covers its own two 64B cachelines.
        f32x4* po = (f32x4*)(out + ((size_t)b << 5));
        #pragma unroll
        for (int q = 0; q < 8; ++q) {
            f32x4 v = {acc[4 * q], acc[4 * q + 1], acc[4 * q + 2], acc[4 * q + 3]};
            __builtin_nontemporal_store(v, po + q);
        }
    }
}

extern "C" void kernel_launch(void* const* d_in, const int* in_sizes, int n_in,
                              void* d_out, int out_size, void* d_ws, size_t ws_size,
                              hipStream_t stream) {
    const float* x   = (const float*)d_in[0];   // [B, 3] fp32
    const float* emb = (const float*)d_in[1];   // [7131240, 2] fp32
    // d_in[2] (offsets) is baked into kOff as compile-time constants.
    float* out = (float*)d_out;                 // [B, 32] fp32
    const int B = in_sizes[0] / 3;

    const int threads = 256;
    const int blocks  = 1024;                   // ~4 points/thread: amortizes TDM stage
    grid_encoder_kernel<<<blocks, threads, 0, stream>>>(x, emb, out, B);
}